// GNN_23716809408524
// MI455X (gfx1250) — compile-verified
//
#include <hip/hip_runtime.h>
#include <hip/hip_bf16.h>

typedef float v2f __attribute__((ext_vector_type(2)));
typedef float v8f __attribute__((ext_vector_type(8)));

#define D 128            // feature width for all hidden layers
#define DOUT 64
#define NGRAPH 128

// ---------------------------------------------------------------------------
// Degree / normalization kernels
// ---------------------------------------------------------------------------
__global__ void init_deg_kernel(float* deg, int n) {
    int i = blockIdx.x * blockDim.x + threadIdx.x;
    if (i < n) deg[i] = 1.0f;                      // self-loop contributes 1
}

__global__ void edge_deg_kernel(const long long* __restrict__ dst, float* deg, int E) {
    int i = blockIdx.x * blockDim.x + threadIdx.x;
    if (i < E) atomicAdd(&deg[(int)dst[i]], 1.0f);
}

__global__ void make_dinv_kernel(float* deg, int n) {   // in-place deg -> rsqrt(deg)
    int i = blockIdx.x * blockDim.x + threadIdx.x;
    if (i < n) {
        float d = deg[i];
        deg[i] = (d > 0.0f) ? rsqrtf(d) : 0.0f;
    }
}

// ---------------------------------------------------------------------------
// GEMM: out[n_rows x 128] = in[n_rows x 128] @ W[128 x 128]
// Pure-fp32 WMMA (V_WMMA_F32_16X16X4_F32). One wave = one 16-row tile,
// 8 waves/block = 128 rows/block. W cached in LDS with even/odd-K interleave
// so each B fragment is one 8-byte LDS load.
// ---------------------------------------------------------------------------
__global__ __launch_bounds__(256) void gemm128_wmma_kernel(
    const float* __restrict__ in, const float* __restrict__ W,
    float* __restrict__ out, int n_rows)
{
    __shared__ float Wlds[D * D];   // 64 KB

    // Cooperative load + swizzle: (k, col) -> (k>>1)*256 + col*2 + (k&1)
    for (int i = threadIdx.x; i < D * D; i += 256) {
        int k = i >> 7, col = i & 127;
        Wlds[(k >> 1) * 256 + col * 2 + (k & 1)] = W[i];
    }
    __syncthreads();

    const int wave = threadIdx.x >> 5;
    const int lane = threadIdx.x & 31;
    const int idx  = lane & 15;       // M (for A) / N (for B,C,D) position
    const int hi   = lane >> 4;       // lane half selects K pair {0,1} vs {2,3}

    const int rowBase = blockIdx.x * 128 + wave * 16;
    int row  = rowBase + idx;
    int rowc = row < n_rows ? row : (n_rows - 1);   // clamp: EXEC must stay all-1s
    const float* arow = in + (size_t)rowc * D;

    v8f acc[8];
    #pragma unroll
    for (int t = 0; t < 8; ++t)
        #pragma unroll
        for (int j = 0; j < 8; ++j) acc[t][j] = 0.0f;

    for (int ks = 0; ks < D / 4; ++ks) {
        int kb = ks * 4 + hi * 2;                       // even K base for this lane half
        float2 av = *(const float2*)(arow + kb);        // A frag: K = kb, kb+1
        v2f a; a[0] = av.x; a[1] = av.y;
        const float* wp = &Wlds[(kb >> 1) * 256 + idx * 2];
        #pragma unroll
        for (int t = 0; t < 8; ++t) {
            float2 bv = *(const float2*)(wp + t * 32);  // B frag: rows kb,kb+1 col t*16+idx
            v2f b; b[0] = bv.x; b[1] = bv.y;
            acc[t] = __builtin_amdgcn_wmma_f32_16x16x4_f32(
                false, a, false, b, (short)0, acc[t], false, false);
        }
    }

    // C/D layout: VGPR j -> row j (lanes 0-15) or row j+8 (lanes 16-31), col = idx
    #pragma unroll
    for (int j = 0; j < 8; ++j) {
        int r = rowBase + hi * 8 + j;
        if (r < n_rows) {
            float* orow = out + (size_t)r * D;
            #pragma unroll
            for (int t = 0; t < 8; ++t) orow[t * 16 + idx] = acc[t][j];
        }
    }
}

// ---------------------------------------------------------------------------
// agg[n][f] = bias[f] + h[n][f] * dinv[n]^2     (self-loop + bias seed)
// ---------------------------------------------------------------------------
__global__ void agg_init_kernel(const float* __restrict__ h,
                                const float* __restrict__ dinv,
                                const float* __restrict__ bias,
                                float* __restrict__ agg, int n)
{
    int i = blockIdx.x * blockDim.x + threadIdx.x;
    int total = n * D;
    if (i < total) {
        int node = i >> 7, f = i & 127;
        float di = dinv[node];
        agg[i] = bias[f] + h[i] * di * di;
    }
}

// ---------------------------------------------------------------------------
// One wave per edge: agg[dst] += h[src] * dinv[src]*dinv[dst]
// ---------------------------------------------------------------------------
__global__ __launch_bounds__(256) void edge_scatter_kernel(
    const long long* __restrict__ src, const long long* __restrict__ dst,
    const float* __restrict__ h, const float* __restrict__ dinv,
    float* __restrict__ agg, int E)
{
    int e = (blockIdx.x * 256 + threadIdx.x) >> 5;
    int lane = threadIdx.x & 31;
    if (e >= E) return;
    int s = (int)src[e];
    int d = (int)dst[e];
    float w = dinv[s] * dinv[d];
    const float* hs = h + (size_t)s * D;
    float* ad = agg + (size_t)d * D;
    #pragma unroll
    for (int j = 0; j < 4; ++j) {
        int f = lane + j * 32;
        atomicAdd(ad + f, hs[f] * w);
    }
}

__global__ void relu_kernel(float* h, int n) {
    int i = blockIdx.x * blockDim.x + threadIdx.x;
    if (i < n * D) h[i] = fmaxf(h[i], 0.0f);
}

// ---------------------------------------------------------------------------
// Pooling
// ---------------------------------------------------------------------------
__global__ void pool_zero_kernel(float* sums, float* cnts) {
    int i = blockIdx.x * blockDim.x + threadIdx.x;
    if (i < NGRAPH * D) sums[i] = 0.0f;
    if (i < NGRAPH)     cnts[i] = 0.0f;
}

__global__ __launch_bounds__(256) void pool_scatter_kernel(
    const float* __restrict__ h, const long long* __restrict__ batch,
    float* sums, float* cnts, int n)
{
    int node = (blockIdx.x * 256 + threadIdx.x) >> 5;
    int lane = threadIdx.x & 31;
    if (node >= n) return;
    int g = (int)batch[node];
    const float* hn = h + (size_t)node * D;
    #pragma unroll
    for (int j = 0; j < 4; ++j)
        atomicAdd(&sums[g * D + lane + j * 32], hn[lane + j * 32]);
    if (lane == 0) atomicAdd(&cnts[g], 1.0f);
}

__global__ void fc_kernel(const float* __restrict__ sums, const float* __restrict__ cnts,
                          const float* __restrict__ Wfc, const float* __restrict__ bfc,
                          float* __restrict__ out)
{
    int g = blockIdx.x;     // 128 graphs
    int o = threadIdx.x;    // 64 outputs
    float inv = 1.0f / fmaxf(cnts[g], 1.0f);
    float acc = bfc[o];
    const float* pg = sums + g * D;
    for (int f = 0; f < D; ++f)
        acc = fmaf(pg[f] * inv, Wfc[f * DOUT + o], acc);
    out[g * DOUT + o] = acc;
}

// ---------------------------------------------------------------------------
// Orchestration
// ---------------------------------------------------------------------------
extern "C" void kernel_launch(void* const* d_in, const int* in_sizes, int n_in,
                              void* d_out, int out_size, void* d_ws, size_t ws_size,
                              hipStream_t stream) {
    const float*      x     = (const float*)d_in[0];
    const long long*  ei    = (const long long*)d_in[1];   // [2, E] int64
    const long long*  batch = (const long long*)d_in[2];
    const float* W1 = (const float*)d_in[3];
    const float* b1 = (const float*)d_in[4];
    const float* W2 = (const float*)d_in[5];
    const float* b2 = (const float*)d_in[6];
    const float* W3 = (const float*)d_in[7];
    const float* b3 = (const float*)d_in[8];
    const float* Wfc = (const float*)d_in[9];
    const float* bfc = (const float*)d_in[10];
    float* out = (float*)d_out;

    const int N = in_sizes[0] / D;       // 100000
    const int E = in_sizes[1] / 2;       // 1600000
    const long long* src = ei;
    const long long* dst = ei + E;

    // Workspace carve-up (all fp32):
    float* ws   = (float*)d_ws;
    size_t NF   = (size_t)N * D;
    float* bufA = ws;                    // N*128
    float* bufB = bufA + NF;             // N*128
    float* bufC = bufB + NF;             // N*128
    float* dinv = bufC + NF;             // N
    float* sums = dinv + N;              // 128*128
    float* cnts = sums + NGRAPH * D;     // 128

    const int gN    = (N + 255) / 256;
    const int gE    = (E + 255) / 256;
    const int gGemm = (N + 127) / 128;
    const int gElem = (int)((NF + 255) / 256);
    const int gEdge = (E + 7) / 8;       // 8 edges (waves) per 256-thread block
    const int gNode = (N + 7) / 8;       // 8 nodes per block for pooling

    // Degrees -> dinv = rsqrt(deg) with self loops
    init_deg_kernel<<<gN, 256, 0, stream>>>(dinv, N);
    edge_deg_kernel<<<gE, 256, 0, stream>>>(dst, dinv, E);
    make_dinv_kernel<<<gN, 256, 0, stream>>>(dinv, N);

    // Layer 1: h = x @ W1 (bufA); agg = bufB
    gemm128_wmma_kernel<<<gGemm, 256, 0, stream>>>(x, W1, bufA, N);
    agg_init_kernel<<<gElem, 256, 0, stream>>>(bufA, dinv, b1, bufB, N);
    edge_scatter_kernel<<<gEdge, 256, 0, stream>>>(src, dst, bufA, dinv, bufB, E);
    relu_kernel<<<gElem, 256, 0, stream>>>(bufB, N);

    // Layer 2: in = bufB; h = bufA; agg = bufC
    gemm128_wmma_kernel<<<gGemm, 256, 0, stream>>>(bufB, W2, bufA, N);
    agg_init_kernel<<<gElem, 256, 0, stream>>>(bufA, dinv, b2, bufC, N);
    edge_scatter_kernel<<<gEdge, 256, 0, stream>>>(src, dst, bufA, dinv, bufC, E);
    relu_kernel<<<gElem, 256, 0, stream>>>(bufC, N);

    // Layer 3: in = bufC; h = bufB; agg = bufA
    gemm128_wmma_kernel<<<gGemm, 256, 0, stream>>>(bufC, W3, bufB, N);
    agg_init_kernel<<<gElem, 256, 0, stream>>>(bufB, dinv, b3, bufA, N);
    edge_scatter_kernel<<<gEdge, 256, 0, stream>>>(src, dst, bufB, dinv, bufA, E);
    relu_kernel<<<gElem, 256, 0, stream>>>(bufA, N);

    // Global mean pool + FC
    pool_zero_kernel<<<(NGRAPH * D + 255) / 256, 256, 0, stream>>>(sums, cnts);
    pool_scatter_kernel<<<gNode, 256, 0, stream>>>(bufA, batch, sums, cnts, N);
    fc_kernel<<<NGRAPH, DOUT, 0, stream>>>(sums, cnts, Wfc, bfc, out);
}